// BaseSurpriseRouter_73031623901313
// MI455X (gfx1250) — compile-verified
//
#include <hip/hip_runtime.h>
#include <cstdint>
#include <cstddef>

// Problem constants (from reference): B=4, T=8192, D=1024, k = T/2 = 4096
#define TB 4
#define TT 8192
#define TD 1024
#define KSEL 4096
#define KL_C 1e-8f
#define BETA_CE 10.0f

typedef float f4  __attribute__((ext_vector_type(4)));
typedef float v2f __attribute__((ext_vector_type(2)));
typedef float v8f __attribute__((ext_vector_type(8)));

// ---------------------------------------------------------------------------
// Kernel 1: per-(b,t) KL scores -> S_CE, streaming 5 x [B,T,D] f32 (NT loads).
// One wave per row. Cross-lane reduction of the two accumulators is done with
// a single V_WMMA_F32_16X16X4_F32 (dual-channel lane-pair reduction).
// ---------------------------------------------------------------------------
__global__ __launch_bounds__(256) void surprise_score_kernel(
    const float* __restrict__ mu_p,
    const float* __restrict__ mu_q,
    const float* __restrict__ log_var_q,
    const float* __restrict__ mu_q_ch,
    const float* __restrict__ log_var_q_ch,
    const float* __restrict__ raw_o_ce,
    float* __restrict__ scores)
{
    const int wave = threadIdx.x >> 5;
    const int lane = threadIdx.x & 31;
    const int row  = blockIdx.x * 8 + wave;        // 0 .. B*T-1 (grid sized exactly)
    const size_t base = (size_t)row * TD;

    const f4* mp4 = reinterpret_cast<const f4*>(mu_p        + base);
    const f4* mq4 = reinterpret_cast<const f4*>(mu_q        + base);
    const f4* lv4 = reinterpret_cast<const f4*>(log_var_q   + base);
    const f4* mc4 = reinterpret_cast<const f4*>(mu_q_ch     + base);
    const f4* lc4 = reinterpret_cast<const f4*>(log_var_q_ch+ base);

    float acc_st = 0.0f;   // sum over this lane's 32 features: lv + ((dmu)^2+c)*exp(-lv)
    float acc_ch = 0.0f;

#pragma unroll
    for (int q = 0; q < 8; ++q) {
        const int i = (q << 5) + lane;             // coalesced 512B/wave per array
        f4 mp = __builtin_nontemporal_load(mp4 + i);
        f4 mq = __builtin_nontemporal_load(mq4 + i);
        f4 lv = __builtin_nontemporal_load(lv4 + i);
        f4 mc = __builtin_nontemporal_load(mc4 + i);
        f4 lc = __builtin_nontemporal_load(lc4 + i);
#pragma unroll
        for (int c = 0; c < 4; ++c) {
            float d  = mp[c] - mq[c];
            acc_st  += lv[c] + (d  * d  + KL_C) * __expf(-lv[c]);
            float dc = mp[c] - mc[c];
            acc_ch  += lc[c] + (dc * dc + KL_C) * __expf(-lc[c]);
        }
    }

    // --- WMMA dual-channel reduction -------------------------------------
    // A (16x4 f32): lanes 0-15 -> {K0=acc_st, K1=acc_ch}; lanes 16-31 -> {K2,K3}.
    // B (4x16 f32): rows {0,2} = 1 for cols 0-7; rows {1,3} = 1 for cols 8-15.
    // => D[M, n<8] = st_M + st_{M+16};  D[M, n>=8] = ch_M + ch_{M+16}.
    v2f a; a[0] = acc_st; a[1] = acc_ch;
    v2f b; b[0] = ((lane & 15) < 8) ? 1.0f : 0.0f; b[1] = 1.0f - b[0];
    v8f c{};
    c = __builtin_amdgcn_wmma_f32_16x16x4_f32(false, a, false, b,
                                              (short)0, c, false, false);
    // Each lane holds D rows {j+8h} of its column: sum them (8 VALU adds),
    // then fold the two half-wave row groups with one xor-shuffle.
    float local = c[0] + c[1] + c[2] + c[3] + c[4] + c[5] + c[6] + c[7];
    local += __shfl_xor(local, 16, 32);
    // lanes {0-7,16-23} now hold the full st sum; {8-15,24-31} the ch sum.
    const float inv2d = 0.5f / (float)TD;
    float d_st = __shfl(local, 0, 32) * inv2d;
    float d_ch = __shfl(local, 8, 32) * inv2d;

    float ce = d_st - d_ch + __logf(raw_o_ce[0] + 1e-10f);
    float s  = 1.0f / (1.0f + __expf(-BETA_CE * ce));
    if (lane == 0) scores[row] = s;
}

// ---------------------------------------------------------------------------
// Kernel 2: per-batch sorted top-k via in-LDS bitonic sort of 8192 u64 keys.
// key = (monotonic(val) << 32) | (8191 - idx): descending sort == descending
// values with lower-index-first tie break (jax.lax.top_k semantics).
// ---------------------------------------------------------------------------
__device__ __forceinline__ unsigned map_f32(float f) {
    unsigned u = __float_as_uint(f);
    return (u & 0x80000000u) ? ~u : (u | 0x80000000u);
}
__device__ __forceinline__ float unmap_f32(unsigned m) {
    unsigned u = (m & 0x80000000u) ? (m & 0x7FFFFFFFu) : ~m;
    return __uint_as_float(u);
}

__global__ __launch_bounds__(1024) void topk_sort_kernel(
    const float* __restrict__ scores,
    int* __restrict__ topk_idx,
    float* __restrict__ topk_val)
{
    extern __shared__ unsigned long long keys[];     // 8192 * 8B = 64 KB LDS
    const int b = blockIdx.x;
    const float* s = scores + (size_t)b * TT;

    for (int i = threadIdx.x; i < TT; i += 1024) {
        keys[i] = ((unsigned long long)map_f32(s[i]) << 32)
                | (unsigned)(TT - 1 - i);
    }
    __syncthreads();

    for (int k = 2; k <= TT; k <<= 1) {
        for (int j = k >> 1; j > 0; j >>= 1) {
            for (int i = threadIdx.x; i < TT; i += 1024) {
                const int ixj = i ^ j;
                if (ixj > i) {
                    unsigned long long x = keys[i], y = keys[ixj];
                    const bool desc = ((i & k) == 0);
                    if (desc ? (x < y) : (x > y)) { keys[i] = y; keys[ixj] = x; }
                }
            }
            __syncthreads();
        }
    }

    for (int i = threadIdx.x; i < KSEL; i += 1024) {
        unsigned long long key = keys[i];
        topk_idx[b * KSEL + i] = TT - 1 - (int)(key & 0xFFFFFFFFull);
        topk_val[b * KSEL + i] = unmap_f32((unsigned)(key >> 32));
    }
}

// ---------------------------------------------------------------------------
// Kernel 3: gather selected rows of hidden_states and scale by gate value.
// One block per output row; one float4 per thread; NT stores (write-once).
// ---------------------------------------------------------------------------
__global__ __launch_bounds__(256) void gather_scale_kernel(
    const float* __restrict__ hidden,
    const int* __restrict__ topk_idx,
    const float* __restrict__ topk_val,
    float* __restrict__ out)
{
    const int r = blockIdx.x;                 // 0 .. B*KSEL-1
    const int b = r >> 12;                    // r / 4096
    const int t = topk_idx[r];
    const float g = topk_val[r];

    const f4* src = reinterpret_cast<const f4*>(hidden + ((size_t)b * TT + t) * TD);
    f4* dst = reinterpret_cast<f4*>(out + (size_t)r * TD);

    f4 v = src[threadIdx.x];
    v[0] *= g; v[1] *= g; v[2] *= g; v[3] *= g;
    __builtin_nontemporal_store(v, dst + threadIdx.x);
}

// ---------------------------------------------------------------------------
extern "C" void kernel_launch(void* const* d_in, const int* in_sizes, int n_in,
                              void* d_out, int out_size, void* d_ws, size_t ws_size,
                              hipStream_t stream)
{
    (void)in_sizes; (void)n_in; (void)out_size; (void)ws_size;

    const float* mu_p         = (const float*)d_in[0];
    const float* mu_q         = (const float*)d_in[1];
    const float* log_var_q    = (const float*)d_in[2];
    const float* mu_q_ch      = (const float*)d_in[3];
    const float* log_var_q_ch = (const float*)d_in[4];
    const float* hidden       = (const float*)d_in[5];
    const float* raw_o_ce     = (const float*)d_in[6];
    // d_in[7] (raw_m_cu) only feeds CU/S_CU, which never reach the output.

    char* ws = (char*)d_ws;
    float* scores   = (float*)ws;                               // B*T f32   (128 KB)
    int*   topk_idx = (int*)  (ws + (size_t)TB * TT * 4);       // B*k  i32  ( 64 KB)
    float* topk_val = (float*)(ws + (size_t)TB * TT * 4
                                  + (size_t)TB * KSEL * 4);     // B*k  f32  ( 64 KB)

    // Pass 1: scores. 32768 rows, 1 wave/row, 8 waves/block -> 4096 blocks.
    surprise_score_kernel<<<(TB * TT) / 8, 256, 0, stream>>>(
        mu_p, mu_q, log_var_q, mu_q_ch, log_var_q_ch, raw_o_ce, scores);

    // Pass 2: per-batch sorted top-k (bitonic in 64 KB dynamic LDS).
    topk_sort_kernel<<<TB, 1024, TT * sizeof(unsigned long long), stream>>>(
        scores, topk_idx, topk_val);

    // Pass 3: gather + gate-scale into [B*k, D] output.
    gather_scale_kernel<<<TB * KSEL, 256, 0, stream>>>(
        hidden, topk_idx, topk_val, (float*)d_out);
}